// KbNufft_58274116272737
// MI455X (gfx1250) — compile-verified
//
#include <hip/hip_runtime.h>
#include <math.h>

typedef float v2f __attribute__((ext_vector_type(2)));
typedef float v8f __attribute__((ext_vector_type(8)));

#define B_      2
#define C_      16
#define M_      409600
#define IMG     256
#define K0_     512
#define NPIX    (K0_ * K0_)     // 262144 = 2^18
#define TABLEN  6145            // J*L + 1
#define CENTER_ 3072            // J*L/2

// ---------------------------------------------------------------------------
// D = A(16x4) * B(4x16) + C  on the CDNA5 matrix pipe.
// ---------------------------------------------------------------------------
__device__ __forceinline__ v8f wmma_f32(v2f a, v2f b, v8f c) {
    return __builtin_amdgcn_wmma_f32_16x16x4_f32(false, a, false, b,
                                                 (short)0, c, false, false);
}

// ---------------------------------------------------------------------------
// Kernel 1: apodize by scaling_coef and zero-pad 256x256 -> 512x512.
// ---------------------------------------------------------------------------
__global__ __launch_bounds__(256) void apod_pad(const float* __restrict__ x,
                                                const float* __restrict__ sc,
                                                float2* __restrict__ g0) {
    int idx = blockIdx.x * blockDim.x + threadIdx.x;   // over B*C*NPIX
    int pix = idx & (NPIX - 1);
    int bc  = idx >> 18;
    int i0  = pix >> 9;
    int i1  = pix & 511;
    float2 v = make_float2(0.0f, 0.0f);
    if (i0 < IMG && i1 < IMG) {
        int p   = i0 * IMG + i1;
        float xr = x[(bc * 2 + 0) * (IMG * IMG) + p];
        float xi = x[(bc * 2 + 1) * (IMG * IMG) + p];
        float sr = sc[p];
        float si = sc[IMG * IMG + p];
        v.x = xr * sr - xi * si;
        v.y = xr * si + xi * sr;
    }
    g0[idx] = v;
}

// ---------------------------------------------------------------------------
// Kernel 2: one 512-point complex FFT per wave along the contiguous axis,
// result written TRANSPOSED (out[k][row]).  Two passes => full 2-D FFT with
// final layout [k0][k1] (k1 contiguous).
//
// Four-step: n = n2 + 32*n1, k = k1 + 16*k2.
//   X1[k1][n2] = sum_n1 x[n2+32*n1] W16^{n1 k1}      (WMMA: W16 x (16x32))
//   Y = X1 * W512^{k1 n2}                            (VALU twiddle)
//   E/O = DFT16 over even/odd n2                     (WMMA: (16x16) x W16)
//   X[k1+16k2] = E +/- W32^k2 O                      (radix-2 combine)
// W16 is symmetric, so the same per-lane registers serve A- and B-layouts:
//   operand chunk c, lane l: m=l&15, half=l>>4, vgpr j -> W[m][4c+2*half+j].
// ---------------------------------------------------------------------------
#define WPW 8          // waves per workgroup
#define RST 33         // LDS row stride in float2 (16x32 tile, padded)

__global__ __launch_bounds__(256) void fft512_rows(const float2* __restrict__ gin,
                                                   float2* __restrict__ gout) {
    __shared__ float2 lds[WPW][16 * RST];
    const int tid  = threadIdx.x;
    const int lane = tid & 31;
    const int wv   = tid >> 5;
    const int task = blockIdx.x * WPW + wv;          // 32 images * 512 rows
    const int img  = task >> 9;
    const int row  = task & 511;
    const float2* in     = gin  + (size_t)img * NPIX + (size_t)row * 512;
    float2*       outImg = gout + (size_t)img * NPIX;
    float2* X = lds[wv];

    // straight copy of the row: X[n1][n2] = x[n2 + 32*n1]
#pragma unroll
    for (int t = 0; t < 16; ++t)
        X[t * RST + lane] = in[t * 32 + lane];
    __syncthreads();

    const int mRow = lane & 15;
    const int half = lane >> 4;

    // constant W16 = exp(-2*pi*i*m*k/16) operand chunks
    v2f Wr[4], Wi[4], nWi[4];
#pragma unroll
    for (int c = 0; c < 4; ++c) {
        int k0 = 4 * c + 2 * half;
        float s0, c0, s1, c1;
        __sincosf(-0.3926990817f * (float)((mRow * k0) & 15), &s0, &c0);
        __sincosf(-0.3926990817f * (float)((mRow * (k0 + 1)) & 15), &s1, &c1);
        Wr[c].x = c0;  Wr[c].y = c1;
        Wi[c].x = s0;  Wi[c].y = s1;
        nWi[c].x = -s0; nWi[c].y = -s1;
    }

    // ---- Step 1: X1 = W16 * X  (complex 16x16x32) ----
    v8f Dr[2], Di[2];
#pragma unroll
    for (int t = 0; t < 2; ++t) {
        v8f dr = {0, 0, 0, 0, 0, 0, 0, 0};
        v8f di = {0, 0, 0, 0, 0, 0, 0, 0};
#pragma unroll
        for (int c = 0; c < 4; ++c) {
            int r0 = 4 * c + 2 * half;
            float2 b0 = X[r0 * RST + 16 * t + mRow];
            float2 b1 = X[(r0 + 1) * RST + 16 * t + mRow];
            v2f Br, Bi;
            Br.x = b0.x; Br.y = b1.x;
            Bi.x = b0.y; Bi.y = b1.y;
            dr = wmma_f32(Wr[c],  Br, dr);
            dr = wmma_f32(nWi[c], Bi, dr);
            di = wmma_f32(Wr[c],  Bi, di);
            di = wmma_f32(Wi[c],  Br, di);
        }
        Dr[t] = dr; Di[t] = di;
    }
    __syncthreads();

    // ---- Step 2: twiddle W512^{k1*n2}, store Y back to LDS ----
#pragma unroll
    for (int t = 0; t < 2; ++t) {
#pragma unroll
        for (int v = 0; v < 8; ++v) {
            int k1 = v + 8 * half;                    // D row
            int n2 = 16 * t + mRow;                   // D col
            float s, cc;
            __sincosf(-1.22718463e-2f * (float)((k1 * n2) & 511), &s, &cc);
            float yr = Dr[t][v] * cc - Di[t][v] * s;
            float yi = Dr[t][v] * s + Di[t][v] * cc;
            X[k1 * RST + n2] = make_float2(yr, yi);
        }
    }
    __syncthreads();

    // ---- Step 3: E = Ye*W16, O = Yo*W16 (even/odd n2 columns) ----
    v8f Er = {0, 0, 0, 0, 0, 0, 0, 0}, Ei = {0, 0, 0, 0, 0, 0, 0, 0};
    v8f Or = {0, 0, 0, 0, 0, 0, 0, 0}, Oi = {0, 0, 0, 0, 0, 0, 0, 0};
#pragma unroll
    for (int c = 0; c < 4; ++c) {
        int m0 = 4 * c + 2 * half;
        const float2* rp = &X[(lane & 15) * RST + 2 * m0];
        float2 e0 = rp[0], o0 = rp[1], e1 = rp[2], o1 = rp[3];
        v2f Aer, Aei, Aor, Aoi;
        Aer.x = e0.x; Aer.y = e1.x;   Aei.x = e0.y; Aei.y = e1.y;
        Aor.x = o0.x; Aor.y = o1.x;   Aoi.x = o0.y; Aoi.y = o1.y;
        Er = wmma_f32(Aer, Wr[c],  Er);
        Er = wmma_f32(Aei, nWi[c], Er);
        Ei = wmma_f32(Aer, Wi[c],  Ei);
        Ei = wmma_f32(Aei, Wr[c],  Ei);
        Or = wmma_f32(Aor, Wr[c],  Or);
        Or = wmma_f32(Aoi, nWi[c], Or);
        Oi = wmma_f32(Aor, Wi[c],  Oi);
        Oi = wmma_f32(Aoi, Wr[c],  Oi);
    }

    // ---- Step 4: radix-2 combine, transposed store ----
    const int kap = lane & 15;
    float s32, c32;
    __sincosf(-0.1963495408f * (float)kap, &s32, &c32);   // W32^kap
#pragma unroll
    for (int v = 0; v < 8; ++v) {
        int k1 = v + 8 * half;
        float tr = c32 * Or[v] - s32 * Oi[v];
        float ti = c32 * Oi[v] + s32 * Or[v];
        int k = k1 + 16 * kap;
        outImg[(size_t)k * 512 + row]         = make_float2(Er[v] + tr, Ei[v] + ti);
        outImg[(size_t)(k + 256) * 512 + row] = make_float2(Er[v] - tr, Ei[v] - ti);
    }
}

// ---------------------------------------------------------------------------
// Kernel 3: table interpolation.  One thread per (b, m); the 36 separable
// weights (phase folded in) live in VGPRs and are reused across all 16 coils.
// All 36x16 gathers hit the L2-resident 64 MB grid.
// ---------------------------------------------------------------------------
__global__ __launch_bounds__(256) void interp(const float* __restrict__ om,
                                              const float* __restrict__ tab0,
                                              const float* __restrict__ tab1,
                                              const float* __restrict__ nsh,
                                              const float2* __restrict__ grid,
                                              float* __restrict__ out) {
    int gid = blockIdx.x * blockDim.x + threadIdx.x;   // over B_*M_
    int b = gid / M_;
    int m = gid - b * M_;

    float om0 = om[(b * 2 + 0) * M_ + m];
    float om1 = om[(b * 2 + 1) * M_ + m];
    const float SC = 81.48733086305042f;               // 512/(2*pi)
    float tm0 = om0 * SC, tm1 = om1 * SC;
    int koff0 = (int)floorf(tm0 - 3.0f) + 1;
    int koff1 = (int)floorf(tm1 - 3.0f) + 1;

    float2 c0[6], c1[6];
    int rowoff[6], col[6];
#pragma unroll
    for (int j = 0; j < 6; ++j) {
        int g0 = koff0 + j, g1 = koff1 + j;
        int t0 = (int)rintf((tm0 - (float)g0) * 1024.0f) + CENTER_;
        int t1 = (int)rintf((tm1 - (float)g1) * 1024.0f) + CENTER_;
        t0 = min(max(t0, 0), TABLEN - 1);
        t1 = min(max(t1, 0), TABLEN - 1);
        c0[j] = make_float2(tab0[t0], tab0[TABLEN + t0]);
        c1[j] = make_float2(tab1[t1], tab1[TABLEN + t1]);
        rowoff[j] = (g0 & 511) * 512;                  // mod-512 wrap
        col[j]    = (g1 & 511);
    }

    float pa = om0 * nsh[0] + om1 * nsh[1];            // fftshift phase
    float pi_, pr_;
    __sincosf(pa, &pi_, &pr_);

    float2 w[36];
#pragma unroll
    for (int a = 0; a < 6; ++a) {
        float ar = c0[a].x * pr_ - c0[a].y * pi_;
        float ai = c0[a].x * pi_ + c0[a].y * pr_;
#pragma unroll
        for (int bb = 0; bb < 6; ++bb) {
            w[a * 6 + bb].x = ar * c1[bb].x - ai * c1[bb].y;
            w[a * 6 + bb].y = ar * c1[bb].y + ai * c1[bb].x;
        }
    }

    const float2* gbase = grid + (size_t)b * C_ * NPIX;
#pragma unroll 1
    for (int c = 0; c < C_; ++c) {
        const float2* gc = gbase + (size_t)c * NPIX;
        float accR = 0.0f, accI = 0.0f;
#pragma unroll
        for (int a = 0; a < 6; ++a) {
            const float2* rp = gc + rowoff[a];
#pragma unroll
            for (int bb = 0; bb < 6; ++bb) {
                float2 v  = rp[col[bb]];
                float2 ww = w[a * 6 + bb];
                accR += ww.x * v.x - ww.y * v.y;
                accI += ww.x * v.y + ww.y * v.x;
            }
        }
        size_t obase = ((size_t)(b * C_ + c)) * 2 * M_ + m;
        out[obase]      = accR;
        out[obase + M_] = accI;
    }
}

// ---------------------------------------------------------------------------
extern "C" void kernel_launch(void* const* d_in, const int* in_sizes, int n_in,
                              void* d_out, int out_size, void* d_ws, size_t ws_size,
                              hipStream_t stream) {
    const float* x   = (const float*)d_in[0];   // (B,C,2,256,256)
    const float* om  = (const float*)d_in[1];   // (B,2,M)
    const float* sc  = (const float*)d_in[2];   // (2,256,256)
    const float* t0  = (const float*)d_in[3];   // (2,6145)
    const float* t1  = (const float*)d_in[4];   // (2,6145)
    const float* nsh = (const float*)d_in[5];   // (2,)

    float2* g0 = (float2*)d_ws;    // 64 MB grid buffer (B*C*512*512 complex)
    float2* g1 = (float2*)d_out;   // d_out (105 MB) reused as FFT ping buffer;
                                   // fully overwritten by interp afterwards.

    apod_pad   <<< (B_ * C_ * NPIX) / 256, 256, 0, stream >>>(x, sc, g0);
    fft512_rows<<< (B_ * C_ * 512) / WPW, 256, 0, stream >>>(g0, g1); // FFT k1, transpose
    fft512_rows<<< (B_ * C_ * 512) / WPW, 256, 0, stream >>>(g1, g0); // FFT k0, transpose back
    interp     <<< (B_ * M_) / 256, 256, 0, stream >>>(om, t0, t1, nsh, g0, (float*)d_out);
}